// GAT6Model_6124623364714
// MI455X (gfx1250) — compile-verified
//
#include <hip/hip_runtime.h>
#include <math.h>

// ---------------------------------------------------------------------------
// GATv2 x4 + MLP head for MI455X (gfx1250, wave32).
//
// Design notes (MI455X):
//  * All node-feature arrays (50000x64 f32 = 12.8MB each) fit in the 192MB L2,
//    so the per-edge gathers are L2-resident. Wave-per-edge layout turns each
//    gather into coalesced 128B bursts.
//  * Dense GEMMs use V_WMMA_F32_16X16X4_F32 (exact fp32) — GEMM work is only
//    ~2.5 GFLOP total, so fp32 WMMA is free and keeps numerics tight.
//  * Segment-softmax = 3 edge passes: (1) logits + atomic segment max via
//    order-preserving uint encoding, (2) exp + atomic segment sum,
//    (3) normalized weighted scatter with GLOBAL_ATOMIC_ADD_F32.
// ---------------------------------------------------------------------------

typedef __attribute__((ext_vector_type(2))) float v2f;
typedef __attribute__((ext_vector_type(8))) float v8f;

#define WAVE 32

__device__ __forceinline__ float wredsum(float v) {
#pragma unroll
    for (int o = 16; o > 0; o >>= 1) v += __shfl_xor(v, o, WAVE);
    return v;
}

// Order-preserving float<->uint mapping so segment-max can use atomicMax(uint).
__device__ __forceinline__ unsigned encf(float f) {
    unsigned u = __float_as_uint(f);
    return (u & 0x80000000u) ? ~u : (u | 0x80000000u);
}
__device__ __forceinline__ float decf(unsigned u) {
    u = (u & 0x80000000u) ? (u ^ 0x80000000u) : ~u;
    return __uint_as_float(u);
}

// ---------------------------------------------------------------------------
// Y[n, COUT] = X[n, CIN] @ W[CIN, COUT] + bias   via V_WMMA_F32_16X16X4_F32.
// One wave owns a 16-row tile and all COUT columns (COUT/16 accumulators).
// A layout: lanes 0-15 rows M, VGPR{0,1}=K{0,1}; lanes 16-31 VGPR{0,1}=K{2,3}.
// B layout: VGPR0 = rows K0(K2), VGPR1 = rows K1(K3), col = lane&15.
// C/D layout: VGPR v -> M = v + 8*(lane>=16), N = lane&15.
// ---------------------------------------------------------------------------
template<int CIN, int COUT>
__global__ void __launch_bounds__(256)
gemm_bias_kernel(const float* __restrict__ X, const float* __restrict__ W,
                 const float* __restrict__ bias, float* __restrict__ Y,
                 int nRowTiles)
{
    constexpr int NT = COUT / 16;
    const int lane   = threadIdx.x & (WAVE - 1);
    const int waveId = threadIdx.x >> 5;
    const int rowTile = blockIdx.x * 8 + waveId;
    if (rowTile >= nRowTiles) return;

    const int row0  = rowTile * 16;
    const int m     = lane & 15;          // row within tile (A) / col within tile (B,C)
    const int khalf = (lane >> 4) << 1;   // 0 for lanes 0-15, 2 for lanes 16-31
    const float* xrow = X + (size_t)(row0 + m) * CIN;

    v8f acc[NT] = {};

#pragma unroll
    for (int k0 = 0; k0 < CIN; k0 += 4) {
        v2f a;
        a.x = xrow[k0 + khalf];
        a.y = xrow[k0 + khalf + 1];
#pragma unroll
        for (int t = 0; t < NT; ++t) {
            const int col = t * 16 + m;
            v2f b;
            b.x = W[(size_t)(k0 + khalf) * COUT + col];
            b.y = W[(size_t)(k0 + khalf + 1) * COUT + col];
            acc[t] = __builtin_amdgcn_wmma_f32_16x16x4_f32(
                false, a, false, b, (short)0, acc[t], false, false);
        }
    }

    const int rbase = row0 + ((lane >> 4) << 3);  // + v below gives output row
#pragma unroll
    for (int t = 0; t < NT; ++t) {
        const int col = t * 16 + m;
        const float bv = bias[col];
#pragma unroll
        for (int v = 0; v < 8; ++v) {
            Y[(size_t)(rbase + v) * COUT + col] = acc[t][v] + bv;
        }
    }
}

// ---------------------------------------------------------------------------
// Pass 1: per edge, m = leaky_relu(xl[src]+xr[dst]+ew*We0+be), alpha = m.att
//         store alpha, segment-max into amaxi[dst] (encoded uint atomicMax).
// Wave-per-edge: 32 lanes cover C channels (C/32 each), shuffle-reduce dot.
// ---------------------------------------------------------------------------
template<int C>
__global__ void __launch_bounds__(256)
edge_attn_kernel(const float* __restrict__ xl, const float* __restrict__ xr,
                 const int* __restrict__ src, const int* __restrict__ dst,
                 const float* __restrict__ ew,
                 const float* __restrict__ We0, const float* __restrict__ be,
                 const float* __restrict__ att,
                 float* __restrict__ alphaArr, unsigned* __restrict__ amaxi,
                 int E)
{
    const int lane  = threadIdx.x & (WAVE - 1);
    const int wave  = (blockIdx.x * blockDim.x + threadIdx.x) >> 5;
    const int nWave = (gridDim.x * blockDim.x) >> 5;

    for (int e = wave; e < E; e += nWave) {
        const int s = src[e];
        const int d = dst[e];
        const float w = ew[e];
        float partial = 0.f;
#pragma unroll
        for (int i = 0; i < C; i += WAVE) {
            const int c = i + lane;
            float v = xl[(size_t)s * C + c] + xr[(size_t)d * C + c]
                      + w * We0[c] + be[c];
            v = (v > 0.f) ? v : 0.2f * v;     // leaky_relu(0.2)
            partial += v * att[c];
        }
        const float alpha = wredsum(partial);
        if (lane == 0) {
            alphaArr[e] = alpha;
            atomicMax(&amaxi[d], encf(alpha));
        }
    }
}

// ---------------------------------------------------------------------------
// Pass 2: ex = exp(alpha - amax[dst]); denom[dst] += ex. Thread-per-edge.
// ---------------------------------------------------------------------------
__global__ void __launch_bounds__(256)
edge_softmax_kernel(const float* __restrict__ alphaArr,
                    const int* __restrict__ dst,
                    const unsigned* __restrict__ amaxi,
                    float* __restrict__ exv, float* __restrict__ denom, int E)
{
    int t = blockIdx.x * blockDim.x + threadIdx.x;
    const int nT = gridDim.x * blockDim.x;
    for (int e = t; e < E; e += nT) {
        const int d = dst[e];
        const float ex = expf(alphaArr[e] - decf(amaxi[d]));
        exv[e] = ex;
        atomicAdd(&denom[d], ex);
    }
}

// ---------------------------------------------------------------------------
// Pass 3: a = ex/(denom[dst]+1e-16); out[dst] += xl[src] * a. Wave-per-edge.
// ---------------------------------------------------------------------------
template<int C>
__global__ void __launch_bounds__(256)
edge_scatter_kernel(const float* __restrict__ xl,
                    const int* __restrict__ src, const int* __restrict__ dst,
                    const float* __restrict__ exv, const float* __restrict__ denom,
                    float* __restrict__ outb, int E)
{
    const int lane  = threadIdx.x & (WAVE - 1);
    const int wave  = (blockIdx.x * blockDim.x + threadIdx.x) >> 5;
    const int nWave = (gridDim.x * blockDim.x) >> 5;

    for (int e = wave; e < E; e += nWave) {
        const int s = src[e];
        const int d = dst[e];
        const float a = exv[e] / (denom[d] + 1e-16f);
#pragma unroll
        for (int i = 0; i < C; i += WAVE) {
            const int c = i + lane;
            atomicAdd(&outb[(size_t)d * C + c], xl[(size_t)s * C + c] * a);
        }
    }
}

// ---------------------------------------------------------------------------
// Per-node row normalize (mean / std with ddof=1 over 64 channels), with bias.
// Wave per node, 2 channels per lane.
// ---------------------------------------------------------------------------
__global__ void __launch_bounds__(256)
normalize_kernel(const float* __restrict__ outb, const float* __restrict__ bias,
                 float* __restrict__ xnext, int N)
{
    const int lane = threadIdx.x & (WAVE - 1);
    const int node = (blockIdx.x * blockDim.x + threadIdx.x) >> 5;
    if (node >= N) return;
    const float* r = outb + (size_t)node * 64;
    const float v0 = r[lane]      + bias[lane];
    const float v1 = r[lane + 32] + bias[lane + 32];
    const float mu = wredsum(v0 + v1) * (1.f / 64.f);
    const float d0 = v0 - mu, d1 = v1 - mu;
    const float sd = sqrtf(wredsum(d0 * d0 + d1 * d1) * (1.f / 63.f));
    const float inv = 1.f / sd;
    float* w = xnext + (size_t)node * 64;
    w[lane]      = d0 * inv;
    w[lane + 32] = d1 * inv;
}

// ---------------------------------------------------------------------------
// Layer-4 epilogue: colsum[c] += out[node,c] + bias[c]  (C_OUT = 32 = wave).
// ---------------------------------------------------------------------------
__global__ void __launch_bounds__(256)
final_colsum_kernel(const float* __restrict__ outb, const float* __restrict__ bias,
                    float* __restrict__ colsum, int N)
{
    const int lane = threadIdx.x & (WAVE - 1);
    const int node = (blockIdx.x * blockDim.x + threadIdx.x) >> 5;
    if (node >= N) return;
    atomicAdd(&colsum[lane], outb[(size_t)node * 32 + lane] + bias[lane]);
}

// ---------------------------------------------------------------------------
// Head: x5 = colsum/N; x6 = relu(x5@l1_W + l1_b); out = x6@l2_W + l2_b.
// Trivial (32x16 + 16 MACs) -> single thread.
// ---------------------------------------------------------------------------
__global__ void head_kernel(const float* __restrict__ colsum,
                            const float* __restrict__ l1W, const float* __restrict__ l1b,
                            const float* __restrict__ l2W, const float* __restrict__ l2b,
                            float* __restrict__ out, int N)
{
    if (threadIdx.x != 0 || blockIdx.x != 0) return;
    const float invN = 1.f / (float)N;
    float acc = l2b[0];
    for (int j = 0; j < 16; ++j) {
        float s = l1b[j];
        for (int c = 0; c < 32; ++c) s += (colsum[c] * invN) * l1W[c * 16 + j];
        s = (s > 0.f) ? s : 0.f;
        acc += s * l2W[j];
    }
    out[0] = acc;
}

__global__ void __launch_bounds__(256)
zero_kernel(float* __restrict__ p, int n)
{
    int i = blockIdx.x * blockDim.x + threadIdx.x;
    const int st = gridDim.x * blockDim.x;
    for (; i < n; i += st) p[i] = 0.f;
}

// ---------------------------------------------------------------------------
// Host launcher.
// Input order (setup_inputs dict order, params flattened in insertion order):
//   0: features [N,36]   1: edge_index [2,E] (int)   2: edge_weights [E]
//   3+8*l .. : conv{l}: Wl, bl, Wr, br, We, be, att, bias  (l = 0..3)
//   35: l1_W [32,16]  36: l1_b [16]  37: l2_W [16,1]  38: l2_b [1]
// ---------------------------------------------------------------------------
extern "C" void kernel_launch(void* const* d_in, const int* in_sizes, int n_in,
                              void* d_out, int out_size, void* d_ws, size_t ws_size,
                              hipStream_t stream)
{
    (void)n_in; (void)out_size; (void)ws_size;

    const float* features = (const float*)d_in[0];
    const int*   eidx     = (const int*)d_in[1];
    const float* ew       = (const float*)d_in[2];

    const int N = in_sizes[0] / 36;   // 50000
    const int E = in_sizes[2];        // 1600000
    const int* src = eidx;
    const int* dst = eidx + E;

    const float *Wl[4], *bl[4], *Wr[4], *br[4], *We0[4], *be[4], *att[4], *bias[4];
    for (int l = 0; l < 4; ++l) {
        const int b = 3 + 8 * l;
        Wl[l]   = (const float*)d_in[b + 0];
        bl[l]   = (const float*)d_in[b + 1];
        Wr[l]   = (const float*)d_in[b + 2];
        br[l]   = (const float*)d_in[b + 3];
        We0[l]  = (const float*)d_in[b + 4];
        be[l]   = (const float*)d_in[b + 5];
        att[l]  = (const float*)d_in[b + 6];
        bias[l] = (const float*)d_in[b + 7];
    }
    const float* l1W = (const float*)d_in[35];
    const float* l1b = (const float*)d_in[36];
    const float* l2W = (const float*)d_in[37];
    const float* l2b = (const float*)d_in[38];

    // Workspace layout (floats). Total ~77 MB.
    float*    ws     = (float*)d_ws;
    float*    xlb    = ws;                       // N*64
    float*    xrb    = xlb + (size_t)N * 64;     // N*64
    float*    x0     = xrb + (size_t)N * 64;     // N*64
    float*    x1     = x0  + (size_t)N * 64;     // N*64
    unsigned* amaxi  = (unsigned*)(x1 + (size_t)N * 64);   // N   (contig zero region start)
    float*    denom  = (float*)amaxi + N;        // N
    float*    outb   = denom + N;                // N*64  (contig with amaxi/denom)
    float*    alphaA = outb + (size_t)N * 64;    // E
    float*    exv    = alphaA + E;               // E
    float*    colsum = exv + E;                  // 32

    const int rowTiles  = N / 16;                       // 3125 (N multiple of 16)
    const int gemmBlk   = (rowTiles + 7) / 8;
    const int edgeBlk   = 2048;                         // 16384 waves, grid-stride
    const int nodeBlk   = (N + 7) / 8;                  // wave-per-node kernels
    const int zeroN     = N + N + N * 64;               // amaxi + denom + outb

    // ---------------- Layer 1: 36 -> 64, input = features, output -> x0
    zero_kernel<<<512, 256, 0, stream>>>((float*)amaxi, zeroN);
    gemm_bias_kernel<36, 64><<<gemmBlk, 256, 0, stream>>>(features, Wl[0], bl[0], xlb, rowTiles);
    gemm_bias_kernel<36, 64><<<gemmBlk, 256, 0, stream>>>(features, Wr[0], br[0], xrb, rowTiles);
    edge_attn_kernel<64><<<edgeBlk, 256, 0, stream>>>(xlb, xrb, src, dst, ew, We0[0], be[0], att[0], alphaA, amaxi, E);
    edge_softmax_kernel<<<edgeBlk, 256, 0, stream>>>(alphaA, dst, amaxi, exv, denom, E);
    edge_scatter_kernel<64><<<edgeBlk, 256, 0, stream>>>(xlb, src, dst, exv, denom, outb, E);
    normalize_kernel<<<nodeBlk, 256, 0, stream>>>(outb, bias[0], x0, N);

    // ---------------- Layer 2: 64 -> 64, x0 -> x1
    zero_kernel<<<512, 256, 0, stream>>>((float*)amaxi, zeroN);
    gemm_bias_kernel<64, 64><<<gemmBlk, 256, 0, stream>>>(x0, Wl[1], bl[1], xlb, rowTiles);
    gemm_bias_kernel<64, 64><<<gemmBlk, 256, 0, stream>>>(x0, Wr[1], br[1], xrb, rowTiles);
    edge_attn_kernel<64><<<edgeBlk, 256, 0, stream>>>(xlb, xrb, src, dst, ew, We0[1], be[1], att[1], alphaA, amaxi, E);
    edge_softmax_kernel<<<edgeBlk, 256, 0, stream>>>(alphaA, dst, amaxi, exv, denom, E);
    edge_scatter_kernel<64><<<edgeBlk, 256, 0, stream>>>(xlb, src, dst, exv, denom, outb, E);
    normalize_kernel<<<nodeBlk, 256, 0, stream>>>(outb, bias[1], x1, N);

    // ---------------- Layer 3: 64 -> 64, x1 -> x0
    zero_kernel<<<512, 256, 0, stream>>>((float*)amaxi, zeroN);
    gemm_bias_kernel<64, 64><<<gemmBlk, 256, 0, stream>>>(x1, Wl[2], bl[2], xlb, rowTiles);
    gemm_bias_kernel<64, 64><<<gemmBlk, 256, 0, stream>>>(x1, Wr[2], br[2], xrb, rowTiles);
    edge_attn_kernel<64><<<edgeBlk, 256, 0, stream>>>(xlb, xrb, src, dst, ew, We0[2], be[2], att[2], alphaA, amaxi, E);
    edge_softmax_kernel<<<edgeBlk, 256, 0, stream>>>(alphaA, dst, amaxi, exv, denom, E);
    edge_scatter_kernel<64><<<edgeBlk, 256, 0, stream>>>(xlb, src, dst, exv, denom, outb, E);
    normalize_kernel<<<nodeBlk, 256, 0, stream>>>(outb, bias[2], x0, N);

    // ---------------- Layer 4: 64 -> 32, x0 -> colsum -> head
    zero_kernel<<<512, 256, 0, stream>>>((float*)amaxi, zeroN);
    zero_kernel<<<1, 32, 0, stream>>>(colsum, 32);
    gemm_bias_kernel<64, 32><<<gemmBlk, 256, 0, stream>>>(x0, Wl[3], bl[3], xlb, rowTiles);
    gemm_bias_kernel<64, 32><<<gemmBlk, 256, 0, stream>>>(x0, Wr[3], br[3], xrb, rowTiles);
    edge_attn_kernel<32><<<edgeBlk, 256, 0, stream>>>(xlb, xrb, src, dst, ew, We0[3], be[3], att[3], alphaA, amaxi, E);
    edge_softmax_kernel<<<edgeBlk, 256, 0, stream>>>(alphaA, dst, amaxi, exv, denom, E);
    edge_scatter_kernel<32><<<edgeBlk, 256, 0, stream>>>(xlb, src, dst, exv, denom, outb, E);
    final_colsum_kernel<<<nodeBlk, 256, 0, stream>>>(outb, bias[3], colsum, N);
    head_kernel<<<1, 1, 0, stream>>>(colsum, l1W, l1b, l2W, l2b, (float*)d_out, N);
}